// SparseAdagrad_54305566490905
// MI455X (gfx1250) — compile-verified
//
#include <hip/hip_runtime.h>
#include <math.h>
#include <stdint.h>

#define LR   0.01f
#define EPS  1e-10f

// Clang native vector (accepted by __builtin_nontemporal_* and lowers to B128).
typedef __attribute__((ext_vector_type(4))) float v4f;

// ---------------------------------------------------------------------------
// SparseAdagrad on MI455X (gfx1250). Memory-bound: ~1.3 GB traffic @23.3 TB/s
// (~57 us roofline). No matmul exists in this op, so no WMMA; the CDNA5 paths
// used are async-to-LDS B128 loads (ASYNCcnt), global_prefetch_b8, and
// NT temporal hints to keep the 59 MB atomic accumulator resident in the
// 192 MB L2 while 1 GB of stream traffic flows past it.
//
// Inputs  : idx[N] (i32), grad[N*D] (f32), state[V*D] (f32), emb[V*D] (f32)
// Outputs : d_out = [ new_state (V*D f32) | new_emb (V*D f32) ]
// Scratch : d_ws  = count[V] (i32, 4 MB)
// Pipeline (touched-rows-only accumulator traffic; acc lives in out_state):
//   1. zero count[V]
//   2. count[idx[n]]++                      (L2-resident atomics)
//   3. zero acc rows where count>0          (~59 MB instead of 256 MB)
//   4. acc[idx[n]*D+d] += grad[n,d]         (L2-resident float atomics)
//   5. fused update: async-to-LDS stage state/emb (NT), conditional sum read,
//      NT-store both outputs (B128)
// ---------------------------------------------------------------------------

__global__ void k_zero_count(int* __restrict__ count, int V) {
    int tid = blockIdx.x * blockDim.x + threadIdx.x;
    if (tid < V) count[tid] = 0;
}

__global__ void k_count(const int* __restrict__ idx, int* __restrict__ count, int N) {
    int tid = blockIdx.x * blockDim.x + threadIdx.x;
    if (tid < N) atomicAdd(count + idx[tid], 1);
}

__global__ void k_zero_touched(const int* __restrict__ count,
                               v4f* __restrict__ acc4, int total, int logD4) {
    int tid = blockIdx.x * blockDim.x + threadIdx.x;
    if (tid >= total) return;
    int v = tid >> logD4;
    // RT store: these lines are immediately RMW'd by the atomic pass -> keep hot.
    if (count[v] > 0) acc4[tid] = (v4f){0.f, 0.f, 0.f, 0.f};
}

__global__ void k_accum(const int* __restrict__ idx, const float* __restrict__ grad,
                        float* __restrict__ acc, int total, int logD, int Dm1) {
    int tid = blockIdx.x * blockDim.x + threadIdx.x;
    if (tid >= total) return;
    // gfx1250 speculative prefetch of upcoming grad stream (global_prefetch_b8)
    __builtin_prefetch(grad + tid + 16384, 0, 0);
    int n = tid >> logD;
    int d = tid & Dm1;
    int v = idx[n];
    // grad is read exactly once -> NT load; atomics stay RT (L2-resident RMW).
    float g = __builtin_nontemporal_load(grad + tid);
    atomicAdd(acc + ((size_t)v << logD) + (size_t)d, g);
}

__global__ __launch_bounds__(256) void k_final(
        const float* __restrict__ state, const float* __restrict__ emb,
        const int* __restrict__ count,
        v4f* __restrict__ out_state4, v4f* __restrict__ out_emb4,
        int total, int logD4) {
    __shared__ v4f s_state[256];
    __shared__ v4f s_emb[256];

    int tid = blockIdx.x * blockDim.x + threadIdx.x;
    if (tid >= total) return;

    // Byte offsets: per-lane global offset (GVS mode) and LDS destinations.
    unsigned goff  = (unsigned)tid * 16u;                        // <= 256 MB, fits u32
    unsigned lds_s = (unsigned)(uintptr_t)&s_state[threadIdx.x]; // flat low 32 = LDS offset
    unsigned lds_e = (unsigned)(uintptr_t)&s_emb[threadIdx.x];

    // CDNA5 async loads: no VGPR return, tracked on ASYNCcnt. NT: state/emb
    // are streamed once and must not evict the L2-resident accumulator.
    asm volatile("global_load_async_to_lds_b128 %0, %1, %2 th:TH_LOAD_NT"
                 :: "v"(lds_s), "v"(goff), "s"(state) : "memory");
    asm volatile("global_load_async_to_lds_b128 %0, %1, %2 th:TH_LOAD_NT"
                 :: "v"(lds_e), "v"(goff), "s"(emb) : "memory");

    // Overlap: count + (conditional) gradient-sum load while async is in flight.
    int v = tid >> logD4;
    int c = count[v];
    v4f sum = (v4f){0.f, 0.f, 0.f, 0.f};
    if (c > 0) sum = __builtin_nontemporal_load(out_state4 + tid);  // last use

    asm volatile("s_wait_asynccnt 0x0" ::: "memory");
    v4f st = s_state[threadIdx.x];   // ds_load_b128
    v4f em = s_emb[threadIdx.x];     // ds_load_b128

    v4f ns, ne;
    if (c > 0) {
        float inv = 1.0f / (float)c;
        float m0 = sum.x * inv, m1 = sum.y * inv, m2 = sum.z * inv, m3 = sum.w * inv;
        ns.x = st.x + m0 * m0;
        ns.y = st.y + m1 * m1;
        ns.z = st.z + m2 * m2;
        ns.w = st.w + m3 * m3;
        ne.x = em.x - LR * m0 / (sqrtf(ns.x) + EPS);
        ne.y = em.y - LR * m1 / (sqrtf(ns.y) + EPS);
        ne.z = em.z - LR * m2 / (sqrtf(ns.z) + EPS);
        ne.w = em.w - LR * m3 / (sqrtf(ns.w) + EPS);
    } else {
        ns = st;
        ne = em;
    }
    // Outputs are never re-read -> NT stores, don't pollute L2.
    __builtin_nontemporal_store(ns, out_state4 + tid);
    __builtin_nontemporal_store(ne, out_emb4 + tid);
}

extern "C" void kernel_launch(void* const* d_in, const int* in_sizes, int n_in,
                              void* d_out, int out_size, void* d_ws, size_t ws_size,
                              hipStream_t stream) {
    const int*   idx   = (const int*)d_in[0];
    const float* grad  = (const float*)d_in[1];
    const float* state = (const float*)d_in[2];
    const float* emb   = (const float*)d_in[3];

    const int N   = in_sizes[0];               // 262144
    const int ND  = in_sizes[1];               // N * D
    const int VD  = in_sizes[2];               // V * D
    const int D   = ND / N;                    // 64 (power of two)
    const int V   = VD / D;                    // 1,000,000
    const int D4  = D / 4;                     // 16
    const int vd4 = VD / 4;                    // 16M float4 slots per tensor
    const int logD  = __builtin_ctz(D);        // 6
    const int logD4 = __builtin_ctz(D4);       // 4

    float* out_state = (float*)d_out;
    float* out_emb   = out_state + (size_t)VD;
    int*   count     = (int*)d_ws;             // V ints = 4 MB

    const int T = 256;                         // 8 wave32s per block

    k_zero_count<<<(V + T - 1) / T, T, 0, stream>>>(count, V);

    k_count<<<(N + T - 1) / T, T, 0, stream>>>(idx, count, N);

    k_zero_touched<<<(vd4 + T - 1) / T, T, 0, stream>>>(
        count, (v4f*)out_state, vd4, logD4);

    k_accum<<<(ND + T - 1) / T, T, 0, stream>>>(
        idx, grad, out_state, ND, logD, D - 1);

    k_final<<<(vd4 + T - 1) / T, T, 0, stream>>>(
        state, emb, count, (v4f*)out_state, (v4f*)out_emb, vd4, logD4);
}